// mask_80496277061640
// MI455X (gfx1250) — compile-verified
//
#include <hip/hip_runtime.h>
#include <stdint.h>

// Problem constants (fixed by the reference):
//   x, noise: (N=128, L=8192, D=16) float32; P=0.8 -> len_keep = int(8192*0.2) = 1638
//   Outputs concatenated in d_out (float32): x_masked (128*1638*16), mask (128*8192*16),
//   ids_restore (128*8192*16, integer ranks written as exact floats).
#define N_DIM 128
#define L_DIM 8192
#define D_DIM 16
#define LEN_KEEP 1638
#define THREADS 1024

// One workgroup per (n, d) column: 2048 independent stable argsorts of 8192 keys.
// Strategy:
//  1. Async-copy the noise column (stride 64B) from global straight into LDS with
//     global_load_async_to_lds_b32 (CDNA5 async path, tracked by ASYNCcnt), landing each
//     key in the HIGH dword of a u64 slot; a normal LDS store puts the original index
//     in the LOW dword. u64 compare then gives a key-major, index-minor (= stable) order.
//  2. In-place bitonic sort of 8192 u64 in LDS (64 KB of the 320 KB WGP LDS).
//  3. Scatter outputs: rank -> ids_restore / mask; gather x rows for r < len_keep with
//     the D-axis reversal folded into the store address.
__global__ __launch_bounds__(THREADS) void mae_mask_sort_kernel(
    const float* __restrict__ x, const float* __restrict__ noise,
    float* __restrict__ xm_out, float* __restrict__ mask_out,
    float* __restrict__ restore_out)
{
    extern __shared__ unsigned long long lds[];   // 8192 x u64 = 64 KB

    const int tid = threadIdx.x;
    const int bid = blockIdx.x;
    const int n = bid >> 4;      // batch index
    const int d = bid & 15;      // feature column being sorted

    // Base of this (n, :, d) column: element l lives at colBase + l*16 (floats).
    const size_t colBase = (size_t)n * (L_DIM * D_DIM) + d;
    const float* nbase = noise + colBase;

    // ---- Phase 1: fill LDS. Key -> high dword via async global->LDS DMA; index -> low dword.
    for (int l = tid; l < L_DIM; l += THREADS) {
        uint32_t ldsOff = (uint32_t)(uintptr_t)(&lds[l]) + 4u;  // flat addr[31:0] == LDS offset
        uint32_t gOff   = (uint32_t)(l * (D_DIM * 4));          // byte offset within column
        asm volatile("global_load_async_to_lds_b32 %0, %1, %2"
                     :: "v"(ldsOff), "v"(gOff), "s"(nbase)
                     : "memory");
        ((uint32_t*)lds)[2 * l] = (uint32_t)l;                  // low dword = original index
    }
    // Each wave drains its own async counter, then the workgroup barrier covers all waves.
    asm volatile("s_wait_asynccnt 0" ::: "memory");
    __syncthreads();

    // ---- Phase 2: bitonic sort (ascending) of 8192 u64 values.
    // uint32 key-bit ordering == float ordering (keys are non-negative uniforms);
    // low-dword index breaks ties exactly like JAX's stable argsort.
    for (int k = 2; k <= L_DIM; k <<= 1) {
        for (int j = k >> 1; j > 0; j >>= 1) {
            for (int p = tid; p < (L_DIM / 2); p += THREADS) {
                int i       = ((p & ~(j - 1)) << 1) | (p & (j - 1));
                int partner = i | j;
                bool up = ((i & k) == 0);
                unsigned long long a = lds[i];
                unsigned long long b = lds[partner];
                unsigned long long lo = (a < b) ? a : b;
                unsigned long long hi = (a < b) ? b : a;
                lds[i]       = up ? lo : hi;
                lds[partner] = up ? hi : lo;
            }
            __syncthreads();
        }
    }

    // ---- Phase 3: emit outputs.
    // sorted[r] low dword = ids_shuffle[r]; rank r is ids_restore[orig].
    for (int r = tid; r < L_DIM; r += THREADS) {
        unsigned long long v = lds[r];
        int orig = (int)(uint32_t)v;
        size_t gi = colBase + (size_t)orig * D_DIM;       // (n, orig, d) flat index
        restore_out[gi] = (float)r;                       // exact: r < 2^24
        mask_out[gi]    = (r >= LEN_KEEP) ? 1.0f : 0.0f;
        if (r < LEN_KEEP) {
            // x_masked[..., ::-1]: output feature slot is (D-1-d).
            xm_out[(size_t)(n * LEN_KEEP + r) * D_DIM + (D_DIM - 1 - d)] = x[gi];
        }
    }
}

extern "C" void kernel_launch(void* const* d_in, const int* in_sizes, int n_in,
                              void* d_out, int out_size, void* d_ws, size_t ws_size,
                              hipStream_t stream) {
    (void)in_sizes; (void)n_in; (void)out_size; (void)d_ws; (void)ws_size;

    const float* x     = (const float*)d_in[0];
    const float* noise = (const float*)d_in[1];

    float* out     = (float*)d_out;
    float* xm      = out;                                          // 128*1638*16
    float* mask    = xm + (size_t)N_DIM * LEN_KEEP * D_DIM;        // 128*8192*16
    float* restore = mask + (size_t)N_DIM * L_DIM * D_DIM;         // 128*8192*16

    dim3 grid(N_DIM * D_DIM);   // 2048 independent column sorts
    dim3 block(THREADS);
    size_t ldsBytes = (size_t)L_DIM * sizeof(unsigned long long);  // 64 KB

    mae_mask_sort_kernel<<<grid, block, ldsBytes, stream>>>(x, noise, xm, mask, restore);
}